// NT_Xent_pos_Loss_80513456931072
// MI455X (gfx1250) — compile-verified
//
#include <hip/hip_runtime.h>
#include <math.h>

// NT-Xent positive-pair loss: row-wise cosine of 16384 x 512 f32 pairs,
// scalar loss = sum(cos^2) * 2^-27.
//
// Strategy (MI455X, gfx1250, wave32):
//  - Memory-bound (64 MiB in, 4 B out -> ~2.8us at 23.3 TB/s). Keep f32.
//  - Per 16-row tile: dot/normi/normj are diagonals of Gram blocks; compute
//    with V_WMMA_F32_16X16X4_F32 accumulating over K=512.
//  - Wave-private LDS slab (row stride 68 floats: 16B aligned, bank-conflict-
//    free transposed operand reads), coalesced float4 global loads.
//  - Deterministic 2-kernel reduction (no float atomics).

typedef float v2f __attribute__((ext_vector_type(2)));
typedef float v8f __attribute__((ext_vector_type(8)));

#define COLS 512
#define TILE 16
#define CHUNK 64
#define LPAD 68               // 64 + 4 pad floats: bank stride 4 per row
#define WAVES_PER_BLOCK 4
#define NTILES 1024           // 16384 rows / 16

__device__ __forceinline__ float pick8(v8f c, int i) {
  float r = c[0];
  r = (i == 1) ? c[1] : r;
  r = (i == 2) ? c[2] : r;
  r = (i == 3) ? c[3] : r;
  r = (i == 4) ? c[4] : r;
  r = (i == 5) ? c[5] : r;
  r = (i == 6) ? c[6] : r;
  r = (i == 7) ? c[7] : r;
  return r;
}

__global__ __launch_bounds__(WAVES_PER_BLOCK * 32)
void ntxent_pos_tile_kernel(const float* __restrict__ zi,
                            const float* __restrict__ zj,
                            float* __restrict__ partial) {
  __shared__ float lds[WAVES_PER_BLOCK][2][TILE][LPAD];
  const int w    = (int)threadIdx.x >> 5;
  const int lane = (int)threadIdx.x & 31;
  const int tile = (int)blockIdx.x * WAVES_PER_BLOCK + w;
  const int row0 = tile * TILE;

  v8f accIJ = {};  // Zi * Zj^T
  v8f accII = {};  // Zi * Zi^T
  v8f accJJ = {};  // Zj * Zj^T

  // WMMA f32 16x16x4 operand addressing:
  // lanes 0-15  -> row = lane,    K pair (k+0, k+1)
  // lanes 16-31 -> row = lane-16, K pair (k+2, k+3)
  const int idx      = lane & 15;
  const int koff     = (lane >> 4) << 1;
  const int scol     = (lane & 15) * 4;   // staging column (float4 granules)
  const int srow_off = lane >> 4;         // staging row parity

  for (int c = 0; c < COLS / CHUNK; ++c) {
    const int kc = c * CHUNK;
    // Stage 16 rows x 64 cols of each matrix: coalesced b128 loads.
#pragma unroll
    for (int it = 0; it < TILE / 2; ++it) {
      const int r = it * 2 + srow_off;
      const size_t g = (size_t)(row0 + r) * COLS + (size_t)(kc + scol);
      const float4 vi = *reinterpret_cast<const float4*>(zi + g);
      const float4 vj = *reinterpret_cast<const float4*>(zj + g);
      *reinterpret_cast<float4*>(&lds[w][0][r][scol]) = vi;
      *reinterpret_cast<float4*>(&lds[w][1][r][scol]) = vj;
    }
    __builtin_amdgcn_wave_barrier();  // keep staging/compute phases ordered
#pragma unroll
    for (int k = 0; k < CHUNK; k += 4) {
      const float* pi = &lds[w][0][idx][k + koff];
      const float* pj = &lds[w][1][idx][k + koff];
      v2f ai; ai[0] = pi[0]; ai[1] = pi[1];
      v2f aj; aj[0] = pj[0]; aj[1] = pj[1];
      accIJ = __builtin_amdgcn_wmma_f32_16x16x4_f32(false, ai, false, aj,
                                                    (short)0, accIJ, false, false);
      accII = __builtin_amdgcn_wmma_f32_16x16x4_f32(false, ai, false, ai,
                                                    (short)0, accII, false, false);
      accJJ = __builtin_amdgcn_wmma_f32_16x16x4_f32(false, aj, false, aj,
                                                    (short)0, accJJ, false, false);
    }
    __builtin_amdgcn_wave_barrier();  // reads done before next-chunk restage
  }

  // Diagonal (m,m): m<8 -> VGPR m, lane m ; m>=8 -> VGPR m-8, lane m+16.
  const bool valid = (lane < 8) || (lane >= 24);
  const int  vidx  = (lane < 8) ? lane : (lane - 24);
  const float dot  = pick8(accIJ, vidx);
  const float nii  = pick8(accII, vidx);
  const float njj  = pick8(accJJ, vidx);
  const float nprod = sqrtf(nii * njj);
  const float cosv  = dot / fmaxf(nprod, 1e-8f);
  float val = valid ? cosv * cosv : 0.0f;

#pragma unroll
  for (int off = 16; off > 0; off >>= 1) val += __shfl_xor(val, off, 32);
  if (lane == 0) partial[tile] = val;
}

__global__ __launch_bounds__(256)
void ntxent_pos_reduce_kernel(const float* __restrict__ partial,
                              float* __restrict__ out) {
  __shared__ float s[256];
  const int t = (int)threadIdx.x;
  float v = 0.0f;
#pragma unroll
  for (int i = 0; i < NTILES / 256; ++i) v += partial[t + i * 256];
  s[t] = v;
  __syncthreads();
  for (int stride = 128; stride > 0; stride >>= 1) {
    if (t < stride) s[t] += s[t + stride];
    __syncthreads();
  }
  // loss = sum(cos^2) * 8 / N^2, N = 32768 -> scale = 2^-27
  if (t == 0) out[0] = s[0] * 0x1.0p-27f;
}

extern "C" void kernel_launch(void* const* d_in, const int* in_sizes, int n_in,
                              void* d_out, int out_size, void* d_ws, size_t ws_size,
                              hipStream_t stream) {
  (void)in_sizes; (void)n_in; (void)out_size; (void)ws_size;
  const float* zi = (const float*)d_in[0];
  const float* zj = (const float*)d_in[1];
  float* out      = (float*)d_out;
  float* partial  = (float*)d_ws;  // NTILES floats of scratch

  ntxent_pos_tile_kernel<<<NTILES / WAVES_PER_BLOCK, WAVES_PER_BLOCK * 32, 0, stream>>>(
      zi, zj, partial);
  ntxent_pos_reduce_kernel<<<1, 256, 0, stream>>>(partial, out);
}